// Decoder_32796370272640
// MI455X (gfx1250) — compile-verified
//
#include <hip/hip_runtime.h>
#include <hip/hip_bf16.h>
#include <math.h>

typedef __attribute__((ext_vector_type(2))) float v2f;
typedef __attribute__((ext_vector_type(8))) float v8f;

#define BB 64
#define SS 1024
#define HH 1024
#define EE 1024
#define VV 50257
#define G4H 4096          // 4*H
#define KCAT 2048         // E + H  (also = 2*H for [context|h_new])
#define NTILES_OUT ((VV + 15) / 16)   // 3142

__device__ __forceinline__ float sigmoidf_(float x) {
  return 1.0f / (1.0f + __expf(-x));
}

// ---------------------------------------------------------------------------
// K0: build A0 = [ emb[x[b]] | h ]  (64 x 2048), the LHS of the gates GEMM
// ---------------------------------------------------------------------------
__global__ void k_prep(const int* __restrict__ x, const float* __restrict__ h,
                       const float* __restrict__ emb, float* __restrict__ A0) {
  const int b = blockIdx.x;
  const int xb = x[b];                       // x is (B,1)
  for (int i = threadIdx.x; i < HH; i += blockDim.x) {
    A0[b * KCAT + i]      = emb[(size_t)xb * EE + i];
    A0[b * KCAT + EE + i] = h[b * HH + i];
  }
}

// ---------------------------------------------------------------------------
// K1: gates = A0 @ [W_ih | W_hh].T + b_ih + b_hh    (64 x 4096)
// One wave per 16-wide N tile; each wave computes all 4 M tiles so the
// B fragment (weights) is loaded exactly once from DRAM.
// ---------------------------------------------------------------------------
__global__ void k_gates(const float* __restrict__ A0,
                        const float* __restrict__ W_ih,
                        const float* __restrict__ W_hh,
                        const float* __restrict__ b_ih,
                        const float* __restrict__ b_hh,
                        float* __restrict__ gates) {
  const int lane  = threadIdx.x & 31;
  const int wave  = threadIdx.x >> 5;
  const int ntile = blockIdx.x * 8 + wave;   // 0..255
  const int hi    = lane >> 4;               // 0 or 1
  const int col   = lane & 15;               // N (for B,C,D) and M (for A)
  const int n     = ntile * 16 + col;

  v8f acc[4] = {};
  for (int kb = 0; kb < KCAT; kb += 4) {
    const float* Wsel = (kb < EE) ? W_ih : W_hh;
    const int kk = (kb < EE) ? kb : (kb - EE);
    v2f bfrag = *(const v2f*)(Wsel + (size_t)n * EE + kk + 2 * hi);
#pragma unroll
    for (int t = 0; t < 4; ++t) {
      v2f afrag = *(const v2f*)(A0 + (size_t)(t * 16 + col) * KCAT + kb + 2 * hi);
      acc[t] = __builtin_amdgcn_wmma_f32_16x16x4_f32(
          false, afrag, false, bfrag, (short)0, acc[t], false, false);
    }
  }
  const float bias = b_ih[n] + b_hh[n];
#pragma unroll
  for (int t = 0; t < 4; ++t)
#pragma unroll
    for (int g = 0; g < 8; ++g) {
      const int m = t * 16 + g + 8 * hi;
      gates[(size_t)m * G4H + n] = acc[t][g] + bias;
    }
}

// ---------------------------------------------------------------------------
// K2: LSTM cell elementwise -> c_new, h_new; stash h_new into actF[:,H:2H]
// ---------------------------------------------------------------------------
__global__ void k_cell(const float* __restrict__ gates,
                       const float* __restrict__ c_old,
                       float* __restrict__ out_h, float* __restrict__ out_c,
                       float* __restrict__ hn, float* __restrict__ actF) {
  const int idx = blockIdx.x * blockDim.x + threadIdx.x;   // 0..65535
  const int b = idx >> 10, j = idx & 1023;
  const float ig = sigmoidf_(gates[(size_t)b * G4H + j]);
  const float fg = sigmoidf_(gates[(size_t)b * G4H + HH + j]);
  const float gg = tanhf(gates[(size_t)b * G4H + 2 * HH + j]);
  const float og = sigmoidf_(gates[(size_t)b * G4H + 3 * HH + j]);
  const float cn = fg * c_old[idx] + ig * gg;
  const float hv = og * tanhf(cn);
  out_c[idx] = cn;
  out_h[idx] = hv;
  hn[idx] = hv;
  actF[(size_t)b * KCAT + HH + j] = hv;   // second half of [context | h_new]
}

// ---------------------------------------------------------------------------
// K3: q = h_new @ W_bilin   (64 x 1024).  Same WMMA scheme; B is W_bilin[k][n]
// (row-major H x H, NOT transposed), so B fragment needs two strided loads.
// ---------------------------------------------------------------------------
__global__ void k_q(const float* __restrict__ hn,
                    const float* __restrict__ W_bilin,
                    float* __restrict__ q) {
  const int lane  = threadIdx.x & 31;
  const int wave  = threadIdx.x >> 5;
  const int ntile = blockIdx.x * 8 + wave;   // 0..63
  const int hi    = lane >> 4;
  const int col   = lane & 15;
  const int n     = ntile * 16 + col;

  v8f acc[4] = {};
  for (int kb = 0; kb < HH; kb += 4) {
    const int k0 = kb + 2 * hi;
    v2f bfrag;
    bfrag.x = W_bilin[(size_t)k0 * HH + n];
    bfrag.y = W_bilin[(size_t)(k0 + 1) * HH + n];
#pragma unroll
    for (int t = 0; t < 4; ++t) {
      v2f afrag = *(const v2f*)(hn + (size_t)(t * 16 + col) * HH + k0);
      acc[t] = __builtin_amdgcn_wmma_f32_16x16x4_f32(
          false, afrag, false, bfrag, (short)0, acc[t], false, false);
    }
  }
#pragma unroll
  for (int t = 0; t < 4; ++t)
#pragma unroll
    for (int g = 0; g < 8; ++g) {
      const int m = t * 16 + g + 8 * hi;
      q[(size_t)m * HH + n] = acc[t][g];
    }
}

// ---------------------------------------------------------------------------
// K4: online-softmax attention partials (flash-style), encoder read ONCE.
// grid (B, 8 chunks) x 256 thr. Each wave handles 16 s-positions with an
// ASYNC double-buffer: the next encoder row is DMA'd into LDS with
// global_load_async_to_lds_b128 (ASYNCcnt pipe) while the current row is
// consumed from LDS. 8 b128 ops per 4KB row; async loads retire in order,
// so s_wait_asynccnt 8 = "previous row complete, next row still in flight".
// Reference semantics: masked positions get logit 0 (energy*mask) and still
// contribute to softmax and context.
// ---------------------------------------------------------------------------
__global__ void k_attn(const float* __restrict__ q, const float* __restrict__ enc,
                       const int* __restrict__ length,
                       float* __restrict__ pm, float* __restrict__ psum,
                       float* __restrict__ pctx) {
  __shared__ float smem[8 * 2 * 1024];       // 8 waves x double 4KB buffer
  const int b = blockIdx.x;
  const int chunk = blockIdx.y;
  const int lane = threadIdx.x & 31;
  const int wave = threadIdx.x >> 5;
  const int len = length[b];

  // 32-bit LDS byte offset of this wave's buffers (generic LDS addr[31:0]).
  const unsigned lds_wave = (unsigned)(unsigned long long)(&smem[wave * 2048]);

  float qreg[32];
#pragma unroll
  for (int i = 0; i < 32; ++i) qreg[i] = q[(size_t)b * HH + i * 32 + lane];

  float m = -3.0e38f, ssum = 0.0f;
  float acc[32];
#pragma unroll
  for (int i = 0; i < 32; ++i) acc[i] = 0.0f;

  const int s0 = chunk * 128 + wave * 16;

  // issue async copy of enc[b][row] into buffer bufsel (0/1)
#define ASYNC_ROW(row, bufsel)                                                 \
  {                                                                            \
    const unsigned long long gsrc =                                            \
        (unsigned long long)(const void*)(enc + ((size_t)b * SS + (row)) * HH) \
        + (unsigned)(lane * 16);                                               \
    const unsigned ldst = lds_wave + (unsigned)(bufsel) * 4096u +              \
                          (unsigned)(lane * 16);                               \
    _Pragma("unroll")                                                          \
    for (int j_ = 0; j_ < 8; ++j_) {                                           \
      asm volatile("global_load_async_to_lds_b128 %0, %1, off"                 \
                   :: "v"(ldst + (unsigned)(j_ * 512)),                        \
                      "v"(gsrc + (unsigned long long)(j_ * 512))               \
                   : "memory");                                                \
    }                                                                          \
  }

  ASYNC_ROW(s0, 0);                          // prologue: stage row 0

  for (int t = 0; t < 16; ++t) {
    if (t < 15) {
      ASYNC_ROW(s0 + t + 1, (t + 1) & 1);    // prefetch next row
      asm volatile("s_wait_asynccnt 0x8" ::: "memory");   // row t landed
    } else {
      asm volatile("s_wait_asynccnt 0x0" ::: "memory");   // last row landed
    }

    const float* ebuf = &smem[wave * 2048 + (t & 1) * 1024];
    float e[32];
    float dot = 0.0f;
#pragma unroll
    for (int i = 0; i < 32; ++i) {           // conflict-free stride-1 ds reads
      e[i] = ebuf[i * 32 + lane];
      dot = fmaf(e[i], qreg[i], dot);
    }
#pragma unroll
    for (int off = 16; off > 0; off >>= 1)   // wave32 allreduce
      dot += __shfl_xor(dot, off, 32);

    const int s = s0 + t;
    const float logit = (s < len) ? dot : 0.0f;   // energy * mask semantics
    const float mn = fmaxf(m, logit);
    const float alpha = __expf(m - mn);
    const float w = __expf(logit - mn);
#pragma unroll
    for (int i = 0; i < 32; ++i) acc[i] = acc[i] * alpha + w * e[i];
    ssum = ssum * alpha + w;
    m = mn;
  }
#undef ASYNC_ROW

  const int p = b * 64 + chunk * 8 + wave;
  if (lane == 0) { pm[p] = m; psum[p] = ssum; }
#pragma unroll
  for (int i = 0; i < 32; ++i)
    pctx[(size_t)p * HH + i * 32 + lane] = acc[i];
}

// ---------------------------------------------------------------------------
// K5: merge 64 partials per batch -> context, write into actF[:,0:H]
// ---------------------------------------------------------------------------
__global__ void k_combine(const float* __restrict__ pm,
                          const float* __restrict__ psum,
                          const float* __restrict__ pctx,
                          float* __restrict__ actF) {
  __shared__ float wsh[64];
  __shared__ float invZ_s;
  const int b = blockIdx.x;

  if (threadIdx.x == 0) {
    float mstar = -3.0e38f;
    for (int j = 0; j < 64; ++j) mstar = fmaxf(mstar, pm[b * 64 + j]);
    float Z = 0.0f;
    for (int j = 0; j < 64; ++j) {
      const float e = __expf(pm[b * 64 + j] - mstar);
      wsh[j] = e;
      Z += psum[b * 64 + j] * e;
    }
    invZ_s = 1.0f / Z;
  }
  __syncthreads();

  const float invZ = invZ_s;
  for (int hh = threadIdx.x; hh < HH; hh += blockDim.x) {
    float v = 0.0f;
    for (int j = 0; j < 64; ++j)
      v = fmaf(pctx[((size_t)(b * 64 + j)) * HH + hh], wsh[j], v);
    actF[(size_t)b * KCAT + hh] = v * invZ;
  }
}

// ---------------------------------------------------------------------------
// K6: output = [context | h_new] @ W_out.T + b_out   (64 x 50257)
// One wave per N tile, all 4 M tiles per wave => W_out (412 MB) read once.
// ---------------------------------------------------------------------------
__global__ void k_out(const float* __restrict__ actF,
                      const float* __restrict__ W_out,
                      const float* __restrict__ b_out,
                      float* __restrict__ out) {
  const int lane  = threadIdx.x & 31;
  const int wave  = threadIdx.x >> 5;
  const int ntile = blockIdx.x * 8 + wave;
  if (ntile >= NTILES_OUT) return;           // uniform per wave
  const int hi  = lane >> 4;
  const int col = lane & 15;
  const int n   = ntile * 16 + col;
  const int nc  = (n < VV) ? n : (VV - 1);   // clamp keeps EXEC all-ones

  v8f acc[4] = {};
  for (int kb = 0; kb < KCAT; kb += 4) {
    v2f bfrag = *(const v2f*)(W_out + (size_t)nc * KCAT + kb + 2 * hi);
#pragma unroll
    for (int t = 0; t < 4; ++t) {
      v2f afrag = *(const v2f*)(actF + (size_t)(t * 16 + col) * KCAT + kb + 2 * hi);
      acc[t] = __builtin_amdgcn_wmma_f32_16x16x4_f32(
          false, afrag, false, bfrag, (short)0, acc[t], false, false);
    }
  }
  const float bias = b_out[nc];
#pragma unroll
  for (int t = 0; t < 4; ++t)
#pragma unroll
    for (int g = 0; g < 8; ++g) {
      const int m = t * 16 + g + 8 * hi;
      if (n < VV) out[(size_t)m * VV + n] = acc[t][g] + bias;
    }
}

// ---------------------------------------------------------------------------
extern "C" void kernel_launch(void* const* d_in, const int* in_sizes, int n_in,
                              void* d_out, int out_size, void* d_ws, size_t ws_size,
                              hipStream_t stream) {
  const int*   x       = (const int*)d_in[0];
  const float* h       = (const float*)d_in[1];
  const float* c       = (const float*)d_in[2];
  const float* enc     = (const float*)d_in[3];
  const int*   length  = (const int*)d_in[4];
  const float* emb     = (const float*)d_in[5];
  const float* W_ih    = (const float*)d_in[6];
  const float* b_ih    = (const float*)d_in[7];
  const float* W_hh    = (const float*)d_in[8];
  const float* b_hh    = (const float*)d_in[9];
  const float* W_bilin = (const float*)d_in[10];
  const float* W_out   = (const float*)d_in[11];
  const float* b_out   = (const float*)d_in[12];

  float* out   = (float*)d_out;                 // (64, 50257)
  float* out_h = out + (size_t)BB * VV;         // (64, 1024)
  float* out_c = out_h + (size_t)BB * HH;       // (64, 1024)

  float* ws    = (float*)d_ws;
  float* A0    = ws;  ws += (size_t)BB * KCAT;        // 64x2048
  float* gates = ws;  ws += (size_t)BB * G4H;         // 64x4096
  float* hn    = ws;  ws += (size_t)BB * HH;          // 64x1024
  float* q     = ws;  ws += (size_t)BB * HH;          // 64x1024
  float* pm    = ws;  ws += (size_t)BB * 64;          // partial max
  float* psum  = ws;  ws += (size_t)BB * 64;          // partial sum
  float* pctx  = ws;  ws += (size_t)BB * 64 * HH;     // partial context 16MB
  float* actF  = ws;  ws += (size_t)BB * KCAT;        // [context | h_new]

  k_prep<<<BB, 256, 0, stream>>>(x, h, emb, A0);
  k_gates<<<256 / 8, 256, 0, stream>>>(A0, W_ih, W_hh, b_ih, b_hh, gates);
  k_cell<<<(BB * HH) / 256, 256, 0, stream>>>(gates, c, out_h, out_c, hn, actF);
  k_q<<<64 / 8, 256, 0, stream>>>(hn, W_bilin, q);
  dim3 ag(BB, 8);
  k_attn<<<ag, 256, 0, stream>>>(q, enc, length, pm, psum, pctx);
  k_combine<<<BB, 256, 0, stream>>>(pm, psum, pctx, actF);
  k_out<<<(NTILES_OUT + 7) / 8, 256, 0, stream>>>(actF, W_out, b_out, out);
}